// CrossScaleAttention_40114994545135
// MI455X (gfx1250) — compile-verified
//
#include <hip/hip_runtime.h>

typedef __attribute__((ext_vector_type(2))) float v2f;
typedef __attribute__((ext_vector_type(8))) float v8f;

#define D 128
// SCALE = sqrt(D_OUT / NUM_HEADS) = sqrt(16) = 4  ->  multiply by 0.25
#define INV_SCALE 0.25f

// ---- monotone float <-> uint encoding for segment-max via integer atomicMax ----
__device__ __forceinline__ unsigned int fenc(float f) {
    unsigned int u = __float_as_uint(f);
    return (u & 0x80000000u) ? ~u : (u | 0x80000000u);
}
__device__ __forceinline__ float fdec(unsigned int u) {
    return (u & 0x80000000u) ? __uint_as_float(u & 0x7fffffffu)
                             : __uint_as_float(~u);
}
#define ENC_NEG_INF 0x007FFFFFu   // fenc(-inf)

// ---------------------------------------------------------------------------
// GEMM: Y[M,128] = X[M,128] @ W[128,128]^T + b   (torch Linear semantics)
// One 16x16 D-tile per wave via V_WMMA_F32_16X16X4_F32, K-loop of 32 steps.
// Block = 256 threads = 8 waves -> covers all 8 N-tiles of one 16-row M-tile.
// ---------------------------------------------------------------------------
__global__ void __launch_bounds__(256)
qkv_gemm_wmma(const float* __restrict__ X, const float* __restrict__ W,
              const float* __restrict__ bias, float* __restrict__ Y, int M) {
    const int lane = threadIdx.x & 31;
    const int wave = threadIdx.x >> 5;        // N-tile index 0..7
    const int l    = lane & 15;
    const int h    = lane >> 4;               // half-select
    const int m0   = blockIdx.x * 16;
    const int n0   = wave * 16;

    int arowi = m0 + l;
    if (arowi >= M) arowi = M - 1;            // clamp reads (M=50000 is exact anyway)
    const float* arow = X + (size_t)arowi * D;        // A row  (M dimension)
    const float* brow = W + (size_t)(n0 + l) * D;     // W row n -> B column n

    v8f acc = {};
    #pragma unroll 8
    for (int k = 0; k < D; k += 4) {
        const int kb = k + 2 * h;             // lanes 0-15: K=k..k+1, 16-31: K=k+2..k+3
        v2f a = *(const v2f*)(arow + kb);     // A[m][kb], A[m][kb+1]
        v2f b = *(const v2f*)(brow + kb);     // B[kb][n], B[kb+1][n] = W[n][kb..kb+1]
        acc = __builtin_amdgcn_wmma_f32_16x16x4_f32(
            /*neg_a=*/false, a, /*neg_b=*/false, b,
            /*c_mod=*/(short)0, acc, /*reuse_a=*/false, /*reuse_b=*/false);
    }

    const float bn = bias[n0 + l];
    // C/D layout: VGPR r -> row (r + 8*h), col = n0 + l
    float* ybase = Y + (size_t)(m0 + 8 * h) * D + (n0 + l);
    if (m0 + 16 <= M) {
        // wave-uniform fast path: 8 straight-line stores, immediate offsets
        #pragma unroll
        for (int r = 0; r < 8; ++r)
            ybase[(size_t)r * D] = acc[r] + bn;
    } else {
        #pragma unroll
        for (int r = 0; r < 8; ++r) {
            const int m = m0 + r + 8 * h;
            if (m < M) Y[(size_t)m * D + (n0 + l)] = acc[r] + bn;
        }
    }
}

// ---------------------------------------------------------------------------
// Init: zero out[N_dst*128], zero denom, set segment-max to enc(-inf)
// ---------------------------------------------------------------------------
__global__ void init_bufs(unsigned int* __restrict__ mseg, float* __restrict__ denom,
                          float* __restrict__ out, int ndst) {
    const int i = blockIdx.x * blockDim.x + threadIdx.x;
    if (i < ndst) { mseg[i] = ENC_NEG_INF; denom[i] = 0.0f; }
    if (i < ndst * D) out[i] = 0.0f;
}

// ---------------------------------------------------------------------------
// Edge score: one wave per edge. 128-dim dot via float4/lane + shfl reduction,
// then segment-max with integer atomicMax.
// ---------------------------------------------------------------------------
__global__ void __launch_bounds__(256)
edge_score(const float* __restrict__ Q, const float* __restrict__ K,
           const int* __restrict__ src_idx, const int* __restrict__ dst_idx,
           float* __restrict__ score, unsigned int* __restrict__ mseg, int E) {
    const int e    = blockIdx.x * (blockDim.x >> 5) + (threadIdx.x >> 5);
    const int lane = threadIdx.x & 31;
    if (e >= E) return;
    const int s = src_idx[e];
    const int d = dst_idx[e];

    const float4 q = ((const float4*)(Q + (size_t)d * D))[lane];
    const float4 k = ((const float4*)(K + (size_t)s * D))[lane];
    float dot = q.x * k.x + q.y * k.y + q.z * k.z + q.w * k.w;

    #pragma unroll
    for (int off = 16; off > 0; off >>= 1)
        dot += __shfl_xor(dot, off, 32);

    if (lane == 0) {
        const float sc = dot * INV_SCALE;
        score[e] = sc;
        atomicMax(mseg + d, fenc(sc));
    }
}

// ---------------------------------------------------------------------------
// e = exp(score - m[dst]); denom[dst] += e   (one thread per edge)
// score buffer is overwritten in place with e.
// ---------------------------------------------------------------------------
__global__ void __launch_bounds__(256)
edge_exp(float* __restrict__ score, const int* __restrict__ dst_idx,
         const unsigned int* __restrict__ mseg, float* __restrict__ denom, int E) {
    const int e = blockIdx.x * blockDim.x + threadIdx.x;
    if (e >= E) return;
    const int d = dst_idx[e];
    const float ex = __expf(score[e] - fdec(mseg[d]));
    score[e] = ex;
    atomicAdd(denom + d, ex);
}

// ---------------------------------------------------------------------------
// out[dst] += (e/denom[dst]) * V[src]   (one wave per edge, float4 per lane)
// ---------------------------------------------------------------------------
__global__ void __launch_bounds__(256)
edge_scatter(const float* __restrict__ V, const float* __restrict__ expv,
             const int* __restrict__ src_idx, const int* __restrict__ dst_idx,
             const float* __restrict__ denom, float* __restrict__ out, int E) {
    const int e    = blockIdx.x * (blockDim.x >> 5) + (threadIdx.x >> 5);
    const int lane = threadIdx.x & 31;
    if (e >= E) return;
    const int s = src_idx[e];
    const int d = dst_idx[e];
    const float attn = expv[e] / denom[d];

    const float4 v = ((const float4*)(V + (size_t)s * D))[lane];
    float* o = out + (size_t)d * D + lane * 4;
    atomicAdd(o + 0, attn * v.x);
    atomicAdd(o + 1, attn * v.y);
    atomicAdd(o + 2, attn * v.z);
    atomicAdd(o + 3, attn * v.w);
}

// ---------------------------------------------------------------------------
extern "C" void kernel_launch(void* const* d_in, const int* in_sizes, int n_in,
                              void* d_out, int out_size, void* d_ws, size_t ws_size,
                              hipStream_t stream) {
    const float* src_feat = (const float*)d_in[0];   // [N_src, 128]
    const float* dst_feat = (const float*)d_in[1];   // [N_dst, 128]
    const int*   src_idx  = (const int*)d_in[2];     // [E]
    const int*   dst_idx  = (const int*)d_in[3];     // [E]
    const float* Wq = (const float*)d_in[4];
    const float* bq = (const float*)d_in[5];
    const float* Wk = (const float*)d_in[6];
    const float* bk = (const float*)d_in[7];
    const float* Wv = (const float*)d_in[8];
    const float* bv = (const float*)d_in[9];
    float* out = (float*)d_out;

    const int nsrc = in_sizes[0] / D;
    const int ndst = in_sizes[1] / D;
    const int E    = in_sizes[2];

    // workspace carve-out (Q 25.6MB + K 25.6MB + V 25.6MB + score 6.4MB + 2*0.2MB)
    size_t off = 0;
    char* base = (char*)d_ws;
    auto carve = [&](size_t bytes) -> void* {
        void* p = base + off;
        off += (bytes + 255) & ~(size_t)255;
        return p;
    };
    float*        Q     = (float*)carve((size_t)ndst * D * sizeof(float));
    float*        Kf    = (float*)carve((size_t)nsrc * D * sizeof(float));
    float*        Vf    = (float*)carve((size_t)nsrc * D * sizeof(float));
    float*        score = (float*)carve((size_t)E * sizeof(float));
    unsigned int* mseg  = (unsigned int*)carve((size_t)ndst * sizeof(unsigned int));
    float*        denom = (float*)carve((size_t)ndst * sizeof(float));

    // 1) projections via f32 WMMA
    qkv_gemm_wmma<<<(ndst + 15) / 16, 256, 0, stream>>>(dst_feat, Wq, bq, Q,  ndst);
    qkv_gemm_wmma<<<(nsrc + 15) / 16, 256, 0, stream>>>(src_feat, Wk, bk, Kf, nsrc);
    qkv_gemm_wmma<<<(nsrc + 15) / 16, 256, 0, stream>>>(src_feat, Wv, bv, Vf, nsrc);

    // 2) init accumulators / output
    init_bufs<<<(ndst * D + 255) / 256, 256, 0, stream>>>(mseg, denom, out, ndst);

    // 3) edge scores + segment max (wave per edge -> 8 edges per 256-thread block)
    edge_score<<<(E + 7) / 8, 256, 0, stream>>>(Q, Kf, src_idx, dst_idx, score, mseg, E);

    // 4) exp + segment sum
    edge_exp<<<(E + 255) / 256, 256, 0, stream>>>(score, dst_idx, mseg, denom, E);

    // 5) weighted scatter
    edge_scatter<<<(E + 7) / 8, 256, 0, stream>>>(Vf, score, src_idx, dst_idx, denom, out, E);
}